// gcn_65060164600241
// MI455X (gfx1250) — compile-verified
//
#include <hip/hip_runtime.h>

typedef __attribute__((ext_vector_type(2))) float v2f;
typedef __attribute__((ext_vector_type(8))) float v8f;

#define D_IN   128
#define D_H1   16
#define D_H2   128
#define N_CLS  2
#define NGRAPH 128

// ---------------- init: deg=1 (self loop), psum=0, cnt=0 ----------------
__global__ void k_init(float* __restrict__ deg, float* __restrict__ psum,
                       float* __restrict__ cnt, int N) {
  int i = blockIdx.x * blockDim.x + threadIdx.x;
  if (i < N) deg[i] = 1.0f;
  if (i < NGRAPH * D_H2) psum[i] = 0.0f;
  if (i < NGRAPH) cnt[i] = 0.0f;
}

// ---------------- degree accumulation over edges ----------------
__global__ void k_deg(const long long* __restrict__ ei, float* __restrict__ deg, int E) {
  int e = blockIdx.x * blockDim.x + threadIdx.x;
  if (e < E) {
    int d = (int)ei[(size_t)E + e];
    atomicAdd(&deg[d], 1.0f);
  }
}

// ---------------- deg -> rsqrt(deg) in place; also per-graph node counts ----------------
__global__ void k_dinv_cnt(float* __restrict__ deg_dinv, const long long* __restrict__ batch,
                           float* __restrict__ cnt, int N) {
  int n = blockIdx.x * blockDim.x + threadIdx.x;
  if (n < N) {
    deg_dinv[n] = rsqrtf(fmaxf(deg_dinv[n], 1.0f));
    atomicAdd(&cnt[(int)batch[n]], 1.0f);
  }
}

// ---------------- GEMM1: h1 = x @ W1   [N,128]x[128,16] via WMMA f32 16x16x4 ----------------
// one wave per 16-node tile; K=128 -> 32 WMMA steps
__global__ void k_gemm1(const float* __restrict__ x, const float* __restrict__ W1,
                        float* __restrict__ h1, int ntiles) {
  int wave = threadIdx.x >> 5;
  int tile = blockIdx.x * (blockDim.x >> 5) + wave;
  if (tile >= ntiles) return;                 // wave-uniform: EXEC stays all-1s
  int lane = threadIdx.x & 31;
  int rc   = lane & 15;                       // A: row M; B: col N
  int hi   = lane >> 4;                       // 0 -> K pair {0,1}; 1 -> K pair {2,3}
  int m0   = tile * 16;
  const float* xr = x + (size_t)(m0 + rc) * D_IN;
  v8f acc = {};
  #pragma unroll 8
  for (int k = 0; k < D_IN; k += 4) {
    int kk = k + 2 * hi;
    v2f a; a.x = xr[kk];                a.y = xr[kk + 1];
    v2f b; b.x = W1[kk * D_H1 + rc];    b.y = W1[(kk + 1) * D_H1 + rc];
    acc = __builtin_amdgcn_wmma_f32_16x16x4_f32(false, a, false, b, (short)0, acc,
                                                false, false);
  }
  // D layout: VGPR r -> lanes 0-15: M=r, N=lane; lanes 16-31: M=r+8, N=lane-16
  #pragma unroll
  for (int r = 0; r < 8; ++r) {
    int node = m0 + r + 8 * hi;
    h1[(size_t)node * D_H1 + rc] = acc[r];
  }
}

// ---------------- agg init with self-loop term: agg = dinv^2 * h ----------------
__global__ void k_agg_init(const float* __restrict__ h, const float* __restrict__ dinv,
                           float* __restrict__ agg, int N) {
  int i = blockIdx.x * blockDim.x + threadIdx.x;
  if (i < N * D_H1) {
    float di = dinv[i >> 4];
    agg[i] = di * di * h[i];
  }
}

// ---------------- edge scatter-add in 16-dim space (used for both convs) ----------------
// 16 threads per edge, one feature each; gathers + atomics stay in L2 (6.4MB table)
__global__ void k_edge_agg(const float* __restrict__ h, const long long* __restrict__ ei,
                           const float* __restrict__ dinv, float* __restrict__ agg, int E) {
  long long t = (long long)blockIdx.x * blockDim.x + threadIdx.x;
  if (t >= (long long)E * D_H1) return;
  int e = (int)(t >> 4);
  int f = (int)(t & 15);
  int s = (int)ei[e];
  int d = (int)ei[(size_t)E + e];
  float norm = dinv[s] * dinv[d];
  atomicAdd(&agg[(size_t)d * D_H1 + f], h[(size_t)s * D_H1 + f] * norm);
}

// ---------------- r1 = relu(agg1 + b1); buf_agg becomes agg2's self-loop init ----------------
__global__ void k_bias_relu(float* __restrict__ buf_agg, float* __restrict__ r1,
                            const float* __restrict__ b1, const float* __restrict__ dinv,
                            int N) {
  int i = blockIdx.x * blockDim.x + threadIdx.x;
  if (i < N * D_H1) {
    int f = i & 15;
    float r = fmaxf(buf_agg[i] + b1[f], 0.0f);
    r1[i] = r;
    float di = dinv[i >> 4];
    buf_agg[i] = di * di * r;   // self-loop term of second propagation
  }
}

// ---------------- GEMM2 fused with relu + graph mean-pool numerator ----------------
// h2 = relu(agg2 @ W2 + b2); never materialized: atomicAdd into psum[graph][col]
// block = 8 waves; wave w handles column tile w; K=16 -> 4 WMMA steps
__global__ void k_gemm2_pool(const float* __restrict__ agg2, const float* __restrict__ W2,
                             const float* __restrict__ b2, const long long* __restrict__ batch,
                             float* __restrict__ psum, int ntiles) {
  int tile = blockIdx.x;
  if (tile >= ntiles) return;
  int wave = threadIdx.x >> 5;
  int lane = threadIdx.x & 31;
  int rc   = lane & 15;
  int hi   = lane >> 4;
  int m0   = tile * 16;
  int c0   = wave * 16;
  const float* ar = agg2 + (size_t)(m0 + rc) * D_H1;
  v8f acc = {};
  #pragma unroll
  for (int k = 0; k < D_H1; k += 4) {
    int kk = k + 2 * hi;
    v2f a; a.x = ar[kk];                     a.y = ar[kk + 1];
    v2f b; b.x = W2[kk * D_H2 + c0 + rc];    b.y = W2[(kk + 1) * D_H2 + c0 + rc];
    acc = __builtin_amdgcn_wmma_f32_16x16x4_f32(false, a, false, b, (short)0, acc,
                                                false, false);
  }
  float bias = b2[c0 + rc];
  #pragma unroll
  for (int r = 0; r < 8; ++r) {
    int node = m0 + r + 8 * hi;
    float v = fmaxf(acc[r] + bias, 0.0f);
    int g = (int)batch[node];
    atomicAdd(&psum[g * D_H2 + c0 + rc], v);   // 64KB accumulator, L2-resident
  }
}

// ---------------- pooled mean ----------------
__global__ void k_pool_div(float* __restrict__ psum, const float* __restrict__ cnt) {
  int i = blockIdx.x * blockDim.x + threadIdx.x;
  if (i < NGRAPH * D_H2) psum[i] /= fmaxf(cnt[i >> 7], 1.0f);
}

// ---------------- MLP head + log_softmax (tiny: 4 MFLOP) ----------------
__global__ void k_head(const float* __restrict__ pooled, const float* __restrict__ LW1,
                       const float* __restrict__ Lb1, const float* __restrict__ LW2,
                       const float* __restrict__ Lb2, float* __restrict__ out) {
  int g = threadIdx.x;
  if (g >= NGRAPH) return;
  const float* p = pooled + g * D_H2;
  float z0 = Lb2[0], z1 = Lb2[1];
  for (int j = 0; j < D_H2; ++j) {
    float a = Lb1[j];
    for (int k = 0; k < D_H2; ++k) a = fmaf(p[k], LW1[k * D_H2 + j], a);
    a = fmaxf(a, 0.0f);
    z0 = fmaf(a, LW2[j * N_CLS + 0], z0);
    z1 = fmaf(a, LW2[j * N_CLS + 1], z1);
  }
  float m = fmaxf(z0, z1);
  float lse = m + logf(expf(z0 - m) + expf(z1 - m));
  out[g * N_CLS + 0] = z0 - lse;
  out[g * N_CLS + 1] = z1 - lse;
}

extern "C" void kernel_launch(void* const* d_in, const int* in_sizes, int n_in,
                              void* d_out, int out_size, void* d_ws, size_t ws_size,
                              hipStream_t stream) {
  const float*     x     = (const float*)d_in[0];
  const long long* ei    = (const long long*)d_in[1];   // int64 [2, E]
  const long long* batch = (const long long*)d_in[2];   // int64 [N]
  const float*     W1    = (const float*)d_in[3];
  const float*     b1    = (const float*)d_in[4];
  const float*     W2    = (const float*)d_in[5];
  const float*     b2    = (const float*)d_in[6];
  const float*     LW1   = (const float*)d_in[7];
  const float*     Lb1   = (const float*)d_in[8];
  const float*     LW2   = (const float*)d_in[9];
  const float*     Lb2   = (const float*)d_in[10];
  float* out = (float*)d_out;

  int N = in_sizes[0] / D_IN;     // 100000
  int E = in_sizes[1] / 2;        // 1600000
  int ntiles = N / 16;            // 6250 (N divisible by 16)

  // workspace layout (floats): dinv[N] | cnt[128] | psum[16384] | buf0[N*16] | buf1[N*16]
  float* ws   = (float*)d_ws;
  float* dinv = ws;
  float* cnt  = dinv + N;
  float* psum = cnt + NGRAPH;
  float* buf0 = psum + NGRAPH * D_H2;
  float* buf1 = buf0 + (size_t)N * D_H1;

  dim3 blk(256);
  int gN   = (N + 255) / 256;
  int gE   = (E + 255) / 256;
  int gNF  = (N * D_H1 + 255) / 256;
  int gEF  = (int)(((long long)E * D_H1 + 255) / 256);
  int gPS  = (NGRAPH * D_H2 + 255) / 256;

  k_init     <<<gN, blk, 0, stream>>>(dinv, psum, cnt, N);
  k_deg      <<<gE, blk, 0, stream>>>(ei, dinv, E);
  k_dinv_cnt <<<gN, blk, 0, stream>>>(dinv, batch, cnt, N);

  // conv1: linear (WMMA), self-loop init, edge scatter, bias+relu
  k_gemm1    <<<(ntiles + 7) / 8, blk, 0, stream>>>(x, W1, buf0, ntiles);
  k_agg_init <<<gNF, blk, 0, stream>>>(buf0, dinv, buf1, N);
  k_edge_agg <<<gEF, blk, 0, stream>>>(buf0, ei, dinv, buf1, E);
  k_bias_relu<<<gNF, blk, 0, stream>>>(buf1, buf0, b1, dinv, N);

  // conv2: propagate in 16-dim space (linearity), then W2 (WMMA) fused with relu+pool
  k_edge_agg   <<<gEF, blk, 0, stream>>>(buf0, ei, dinv, buf1, E);
  k_gemm2_pool <<<ntiles, blk, 0, stream>>>(buf1, W2, b2, batch, psum, ntiles);

  k_pool_div <<<gPS, blk, 0, stream>>>(psum, cnt);
  k_head     <<<1, 128, 0, stream>>>(psum, LW1, Lb1, LW2, Lb2, out);
}